// Network_80444737454729
// MI455X (gfx1250) — compile-verified
//
#include <hip/hip_runtime.h>
#include <hip/hip_bf16.h>
#include <math.h>

// ---------------- problem constants (from reference setup_inputs) ------------
#define BB 4
#define AA 1024
#define DD 32
#define KK 5
#define NHEAD 5
#define N1 160                 // W1 output dim = DD * NHEAD
#define M1 (BB * AA * (1 + KK))   // 24576 rows through hnn_layer(W1)
#define MIN_NORM_F 1e-15f
#define BALL_EPS_F 0.004f      // c = 1.0 -> sqrt_c = 1

typedef __attribute__((ext_vector_type(2))) float v2f;
typedef __attribute__((ext_vector_type(8))) float v8f;
typedef __attribute__((ext_vector_type(4))) unsigned int v4u;
typedef __attribute__((ext_vector_type(8))) int v8i;
typedef __attribute__((ext_vector_type(4))) int v4i;

// ---------------- wave32 helpers --------------------------------------------
__device__ __forceinline__ float wave_sum(float v) {
#pragma unroll
  for (int off = 16; off > 0; off >>= 1) v += __shfl_xor(v, off, 32);
  return v;
}

__device__ __forceinline__ float artanh_f(float x) {
  x = fminf(fmaxf(x, -1.0f + 1e-7f), 1.0f - 1e-7f);
  return 0.5f * (log1pf(x) - log1pf(-x));
}
__device__ __forceinline__ float tanh15f(float x) {
  return tanhf(fminf(fmaxf(x, -15.0f), 15.0f));
}

// proj for a dim-32 row held one element per lane
__device__ __forceinline__ float proj32(float v) {
  float n = fmaxf(sqrtf(wave_sum(v * v)), MIN_NORM_F);
  float mxn = 1.0f - BALL_EPS_F;
  return (n > mxn) ? v * (mxn / n) : v;
}
// to_hyp = proj(expmap0(u)), dim-32 row
__device__ __forceinline__ float to_hyp32(float u) {
  float n = fmaxf(sqrtf(wave_sum(u * u)), MIN_NORM_F);
  float e = u * (tanh15f(n) / n);
  return proj32(e);
}

// dim-160 rows: 5 regs/lane, element index = lane + 32*j
__device__ __forceinline__ float norm160(const float (&v)[5]) {
  float s = 0.f;
#pragma unroll
  for (int j = 0; j < 5; ++j) s += v[j] * v[j];
  return fmaxf(sqrtf(wave_sum(s)), MIN_NORM_F);
}
__device__ __forceinline__ void proj160(float (&v)[5]) {
  float n = norm160(v);
  float mxn = 1.0f - BALL_EPS_F;
  if (n > mxn) {
    float f = mxn / n;
#pragma unroll
    for (int j = 0; j < 5; ++j) v[j] *= f;
  }
}

// ---------------- kernel 0: hyperbolic biases (b1 -> 160, b2 -> 32) ---------
__global__ void k_bias(const float* __restrict__ b1, const float* __restrict__ b2,
                       float* __restrict__ hb1, float* __restrict__ hb2) {
  int lane = threadIdx.x & 31;
  float mxn = 1.0f - BALL_EPS_F;
  // hyp_bias1 = proj(expmap0(b1)), dim 160
  float v[5];
#pragma unroll
  for (int j = 0; j < 5; ++j) v[j] = b1[lane + 32 * j];
  float n = norm160(v);
  float sc = tanh15f(n) / n;
#pragma unroll
  for (int j = 0; j < 5; ++j) v[j] *= sc;
  n = norm160(v);
  float f = (n > mxn) ? (mxn / n) : 1.0f;
#pragma unroll
  for (int j = 0; j < 5; ++j) hb1[lane + 32 * j] = v[j] * f;
  // hyp_bias2 = proj(expmap0(b2)), dim 32
  float w = b2[lane];
  float nw = fmaxf(sqrtf(wave_sum(w * w)), MIN_NORM_F);
  w *= tanh15f(nw) / nw;
  nw = fmaxf(sqrtf(wave_sum(w * w)), MIN_NORM_F);
  w *= (nw > mxn) ? (mxn / nw) : 1.0f;
  hb2[lane] = w;
}

// ---------------- kernel 1: feats_t = logmap0(in_feats); H[agent] -----------
__global__ void k_feats(const float* __restrict__ in_feats,
                        float* __restrict__ feats_t, float* __restrict__ H) {
  int row = blockIdx.x * (blockDim.x >> 5) + (threadIdx.x >> 5);  // b*A + a
  int lane = threadIdx.x & 31;
  if (row >= BB * AA) return;
  float x = in_feats[row * DD + lane];
  float n = fmaxf(sqrtf(wave_sum(x * x)), MIN_NORM_F);
  float u = x * (artanh_f(n) / n);       // logmap0
  feats_t[row * DD + lane] = u;
  H[(size_t)(row * 6 + 0) * DD + lane] = to_hyp32(u);  // agent hyp row
}

// ---------------- kernel 2: one-hot gather -> H[neighbor] -------------------
// Dominant pass: reads all 84 MB of in_nei with float4 loads, one wave / row.
__global__ void k_gather(const float* __restrict__ in_nei,
                         const float* __restrict__ feats_t,
                         float* __restrict__ H) {
  int r = blockIdx.x * (blockDim.x >> 5) + (threadIdx.x >> 5);  // 0..B*A*K-1
  int lane = threadIdx.x & 31;
  if (r >= BB * AA * KK) return;
  int b = r / (AA * KK);
  int ak = r % (AA * KK);
  int a = ak / KK, k = ak % KK;
  const float4* p = (const float4*)(in_nei + (size_t)r * AA);
  int idx = -1;
  float wv = 0.f;
#pragma unroll
  for (int i = 0; i < 8; ++i) {  // 8 * 32 lanes * 4 floats = 1024
    if (i + 2 < 8) __builtin_prefetch(p + lane + 32 * (i + 2), 0, 0);
    float4 v = p[lane + 32 * i];
    int base = 4 * (lane + 32 * i);
    if (v.x != 0.f) { idx = base + 0; wv = v.x; }
    if (v.y != 0.f) { idx = base + 1; wv = v.y; }
    if (v.z != 0.f) { idx = base + 2; wv = v.z; }
    if (v.w != 0.f) { idx = base + 3; wv = v.w; }
  }
#pragma unroll
  for (int off = 16; off > 0; off >>= 1) {
    int oi = __shfl_xor(idx, off, 32);
    float ow = __shfl_xor(wv, off, 32);
    if (oi > idx) { idx = oi; wv = ow; }
  }
  if (idx < 0) { idx = 0; wv = 0.f; }
  float u = wv * feats_t[(size_t)(b * AA + idx) * DD + lane];  // neigh row
  H[(size_t)((b * AA + a) * 6 + 1 + k) * DD + lane] = to_hyp32(u);
}

// ---------------- kernel 3: WMMA fp32 GEMM  MX[M1 x 160] = H[M1 x 32]·W1^T --
// V_WMMA_F32_16X16X4_F32, K=32 as 8 steps of 4.
// W1 (160x32 fp32, 20 KB) staged in LDS via the Tensor Data Mover: one wave
// per block issues a single 2D-tile TDM descriptor (D#) DMA, waits TENSORcnt.
__global__ void k_gemm1(const float* __restrict__ H, const float* __restrict__ W1,
                        float* __restrict__ MX) {
  __shared__ __align__(16) float sW[N1 * DD];  // 20 KB
  if ((threadIdx.x >> 5) == 0) {
    unsigned lds_off = (unsigned)(uintptr_t)(void*)&sW[0];
    unsigned long long ga = (unsigned long long)(uintptr_t)W1;
    // D# group 0: count=1 (valid), lds_addr, 57-bit global_addr, type=2
    v4u g0 = { 1u,
               lds_off,
               (unsigned)(ga & 0xFFFFFFFFu),
               (unsigned)((ga >> 32) & 0x1FFFFFFu) | (2u << 30) };
    // D# group 1: workgroup_mask=0, data_size=2 (4B), no barrier/iterate/pad,
    // tensor_dim0=32, tensor_dim1=160, tile_dim0=32, tile_dim1=160,
    // tensor_dim0_stride=32
    v8i g1 = { (int)(2u << 16),    // [17:16] data_size = 4 bytes
               (int)(32u << 16),   // tensor_dim0[15:0] at bits [63:48]
               (int)(160u << 16),  // tensor_dim0 hi=0; tensor_dim1 lo=160
               (int)(32u << 16),   // tensor_dim1 hi=0; tile_dim0=32
               160,                // tile_dim1=160; tile_dim2=0
               32,                 // tensor_dim0_stride lo = 32
               0, 0 };             // stride hi / dim1_stride = 0
    v4i gz = { 0, 0, 0, 0 };
#if __clang_major__ >= 23
    v8i gz8 = { 0, 0, 0, 0, 0, 0, 0, 0 };
    __builtin_amdgcn_tensor_load_to_lds(g0, g1, gz, gz, gz8, 0);
#else
    __builtin_amdgcn_tensor_load_to_lds(g0, g1, gz, gz, 0);
#endif
    __builtin_amdgcn_s_wait_tensorcnt(0);
  }
  __syncthreads();

  int wave = blockIdx.x * (blockDim.x >> 5) + (threadIdx.x >> 5);
  int lane = threadIdx.x & 31;
  const int ntiles = N1 / 16;  // 10
  int mt = wave / ntiles, nt = wave % ntiles;
  if (mt >= M1 / 16) return;
  int m0 = mt * 16, n0 = nt * 16;
  int lh = lane & 15, lv = lane >> 4;
  // A frag: lanes 0-15 hold (M=lh, K=kk..kk+1), lanes 16-31 hold K=kk+2..kk+3
  const float* arow = H + (size_t)(m0 + lh) * DD + 2 * lv;
  // B frag (4x16): B[k][n] = W1[n][k]; same K split across lane halves
  const float* brow = sW + (n0 + lh) * DD + 2 * lv;
  v8f acc = {};
#pragma unroll
  for (int kk = 0; kk < DD; kk += 4) {
    v2f af = *(const v2f*)(arow + kk);
    v2f bf = *(const v2f*)(brow + kk);
    acc = __builtin_amdgcn_wmma_f32_16x16x4_f32(
        /*neg_a=*/false, af, /*neg_b=*/false, bf,
        /*c_mod=*/(short)0, acc, /*reuse_a=*/false, /*reuse_b=*/false);
  }
  // C/D layout: VGPR v -> (M = m0 + v + 8*lv, N = n0 + lh)
  float* out = MX + (size_t)(m0 + 8 * lv) * N1 + n0 + lh;
#pragma unroll
  for (int v = 0; v < 8; ++v) out[(size_t)v * N1] = acc[v];
}

// ---------------- kernel 4: per-row hyperbolic post-GEMM (dim 160) ----------
// res = mobius_matvec tail; hyp_linear; hnn_layer tail; then logmap0 -> G.
// G written in place over MX (row-local).
__global__ void k_post1(const float* __restrict__ H, float* __restrict__ MX,
                        const float* __restrict__ hb1) {
  int row = blockIdx.x * (blockDim.x >> 5) + (threadIdx.x >> 5);
  int lane = threadIdx.x & 31;
  if (row >= M1) return;
  float hx = H[(size_t)row * DD + lane];
  float x_norm = fmaxf(sqrtf(wave_sum(hx * hx)), MIN_NORM_F);

  float m[5];
  float s2 = 0.f;
  int zero = 1;
  float* mrow = MX + (size_t)row * N1;
#pragma unroll
  for (int j = 0; j < 5; ++j) {
    m[j] = mrow[lane + 32 * j];
    s2 += m[j] * m[j];
    zero &= (m[j] == 0.f);
  }
  float mx_norm = fmaxf(sqrtf(wave_sum(s2)), MIN_NORM_F);
  int allzero = __all(zero);
  float scale = tanh15f(mx_norm / x_norm * artanh_f(x_norm)) / mx_norm;
  if (allzero) scale = 0.f;
#pragma unroll
  for (int j = 0; j < 5; ++j) m[j] *= scale;  // mobius_matvec result
  proj160(m);
  // mobius_add(res, hyp_bias1), c = 1
  float bb[5];
  float x2l = 0.f, y2l = 0.f, xyl = 0.f;
#pragma unroll
  for (int j = 0; j < 5; ++j) {
    bb[j] = hb1[lane + 32 * j];
    x2l += m[j] * m[j];
    y2l += bb[j] * bb[j];
    xyl += m[j] * bb[j];
  }
  float x2 = wave_sum(x2l), y2 = wave_sum(y2l), xy = wave_sum(xyl);
  float ca = 1.0f + 2.0f * xy + y2;
  float cb = 1.0f - x2;
  float denom = fmaxf(1.0f + 2.0f * xy + x2 * y2, MIN_NORM_F);
#pragma unroll
  for (int j = 0; j < 5; ++j) m[j] = (ca * m[j] + cb * bb[j]) / denom;
  proj160(m);                       // h = hyp_linear output
  float n = norm160(m);             // xt = relu(logmap0(h))
  float sc = artanh_f(n) / n;
#pragma unroll
  for (int j = 0; j < 5; ++j) m[j] = fmaxf(m[j] * sc, 0.f);
  n = norm160(m);                   // proj(expmap0(xt))
  sc = tanh15f(n) / n;
#pragma unroll
  for (int j = 0; j < 5; ++j) m[j] *= sc;
  proj160(m);                       // hnn_layer output
  n = norm160(m);                   // G = logmap0(...) (head features)
  sc = artanh_f(n) / n;
#pragma unroll
  for (int j = 0; j < 5; ++j) mrow[lane + 32 * j] = m[j] * sc;
}

// ---------------- kernel 5: attention + hnn_layer(W2) -----------------------
__global__ void k_attn(const float* __restrict__ G, const float* __restrict__ W2,
                       const float* __restrict__ hb2, float* __restrict__ out,
                       float* __restrict__ att_out) {
  int ba = blockIdx.x * (blockDim.x >> 5) + (threadIdx.x >> 5);
  int lane = threadIdx.x & 31;  // lane == feature d
  if (ba >= BB * AA) return;
  const float* qrow = G + (size_t)(ba * 6) * N1;
  float qh[5], kv[5][5];
#pragma unroll
  for (int h = 0; h < 5; ++h) qh[h] = qrow[lane * NHEAD + h];
#pragma unroll
  for (int j = 0; j < 5; ++j) {
    const float* kr = G + (size_t)(ba * 6 + 1 + j) * N1;
#pragma unroll
    for (int h = 0; h < 5; ++h) kv[j][h] = kr[lane * NHEAD + h];
  }
  float att[5][5];
#pragma unroll
  for (int h = 0; h < 5; ++h)
#pragma unroll
    for (int j = 0; j < 5; ++j) att[h][j] = wave_sum(qh[h] * kv[j][h]);
  // softmax over j
#pragma unroll
  for (int h = 0; h < 5; ++h) {
    float mm = att[h][0];
#pragma unroll
    for (int j = 1; j < 5; ++j) mm = fmaxf(mm, att[h][j]);
    float s = 0.f;
#pragma unroll
    for (int j = 0; j < 5; ++j) { att[h][j] = expf(att[h][j] - mm); s += att[h][j]; }
    float inv = 1.0f / s;
#pragma unroll
    for (int j = 0; j < 5; ++j) att[h][j] *= inv;
  }
  if (lane < 25) att_out[(size_t)ba * 25 + lane] = att[lane / 5][lane % 5];
  // out[d] = mean_h sum_j att * neigh_head
  float o = 0.f;
#pragma unroll
  for (int h = 0; h < 5; ++h)
#pragma unroll
    for (int j = 0; j < 5; ++j) o += att[h][j] * kv[j][h];
  o *= 0.2f;
  // hnn_layer(to_hyp(o), W2, b2), dim 32
  float x = to_hyp32(o);
  float x_norm = fmaxf(sqrtf(wave_sum(x * x)), MIN_NORM_F);
  float mx = 0.f;
#pragma unroll
  for (int d = 0; d < 32; ++d) mx += W2[lane * 32 + d] * __shfl(x, d, 32);
  float mx_norm = fmaxf(sqrtf(wave_sum(mx * mx)), MIN_NORM_F);
  int allz = __all(mx == 0.f);
  float sc = tanh15f(mx_norm / x_norm * artanh_f(x_norm)) / mx_norm;
  if (allz) sc = 0.f;
  float r = proj32(mx * sc);
  // mobius_add with hyp_bias2
  float bbv = hb2[lane];
  float x2 = wave_sum(r * r), y2 = wave_sum(bbv * bbv), xy = wave_sum(r * bbv);
  float denom = fmaxf(1.0f + 2.0f * xy + x2 * y2, MIN_NORM_F);
  r = ((1.0f + 2.0f * xy + y2) * r + (1.0f - x2) * bbv) / denom;
  r = proj32(r);
  float n = fmaxf(sqrtf(wave_sum(r * r)), MIN_NORM_F);
  r = fmaxf(r * (artanh_f(n) / n), 0.f);       // relu(logmap0)
  n = fmaxf(sqrtf(wave_sum(r * r)), MIN_NORM_F);
  r = r * (tanh15f(n) / n);                    // expmap0
  r = proj32(r);
  out[(size_t)ba * DD + lane] = r;
}

// ---------------- launch -----------------------------------------------------
extern "C" void kernel_launch(void* const* d_in, const int* in_sizes, int n_in,
                              void* d_out, int out_size, void* d_ws, size_t ws_size,
                              hipStream_t stream) {
  const float* in_feats = (const float*)d_in[0];  // (4,1024,32)
  const float* in_nei   = (const float*)d_in[1];  // (4,1024,5,1024)
  const float* W1       = (const float*)d_in[2];  // (160,32)
  const float* b1       = (const float*)d_in[3];  // (160,)
  const float* W2       = (const float*)d_in[4];  // (32,32)
  const float* b2       = (const float*)d_in[5];  // (32,)
  float* out_main = (float*)d_out;                        // (4,1024,32)
  float* out_att  = (float*)d_out + (size_t)BB * AA * DD; // (4,1024,5,5)

  float* ws = (float*)d_ws;
  float* feats_t = ws;                                   // 131072
  float* H  = feats_t + (size_t)BB * AA * DD;            // 786432
  float* MX = H + (size_t)M1 * DD;                       // 3932160 (reused as G)
  float* hb1 = MX + (size_t)M1 * N1;                     // 160
  float* hb2 = hb1 + N1;                                 // 32

  k_bias<<<1, 32, 0, stream>>>(b1, b2, hb1, hb2);
  k_feats<<<(BB * AA) / 8, 256, 0, stream>>>(in_feats, feats_t, H);
  k_gather<<<(BB * AA * KK) / 8, 256, 0, stream>>>(in_nei, feats_t, H);
  k_gemm1<<<((M1 / 16) * (N1 / 16)) / 8, 256, 0, stream>>>(H, W1, MX);
  k_post1<<<M1 / 8, 256, 0, stream>>>(H, MX, hb1);
  k_attn<<<(BB * AA) / 8, 256, 0, stream>>>(MX, W2, hb2, out_main, out_att);
}